// PIQuantumDQN_71622874628361
// MI455X (gfx1250) — compile-verified
//
#include <hip/hip_runtime.h>
#include <hip/hip_bf16.h>

#define PI_F 3.14159265358979323846f

typedef __attribute__((ext_vector_type(2))) float v2f;
typedef __attribute__((ext_vector_type(8))) float v8f;

// Swap lane L <-> L^16 within the wave32 (ds_swizzle group-of-32, xor=0x10, and=0x1f).
__device__ __forceinline__ float swapx16(float x) {
  return __int_as_float(__builtin_amdgcn_ds_swizzle(__float_as_int(x), 0x401F));
}

// ---------------------------------------------------------------------------
// Setup: build the fixed per-layer 32x32 real operator M_l = [[Re,-Im],[Im,Re]]
// of U_l = CNOT_ring * (RX(w_l,0) x RX(w_l,1) x RX(w_l,2) x RX(w_l,3)),
// stored directly in WMMA A-operand per-lane layout: mats[(l*32+lane)*32 + j],
// j = itile*16 + k*2 + p  (itile = D row-tile, k = K-chunk, p = A VGPR in pair).
// ---------------------------------------------------------------------------
__global__ void __launch_bounds__(256) qdqn_setup(const float* __restrict__ w,
                                                  float* __restrict__ mats) {
  int t = blockIdx.x * blockDim.x + threadIdx.x;
  if (t >= 5 * 32 * 32) return;
  int l    = t >> 10;
  int rem  = t & 1023;
  int lane = rem & 31;
  int j    = rem >> 5;
  int itile = (j >> 4) & 1;
  int k     = (j >> 1) & 7;
  int p     = j & 1;
  int m   = itile * 16 + (lane & 15);             // output row of M (0..31)
  int col = 4 * k + p + ((lane >> 4) << 1);       // input col of M (0..31)
  int mr = m & 15,  mi = (m >> 4) & 1;            // (amp index, re/im block)
  int cr = col & 15, ci = (col >> 4) & 1;

  // Inverse of the CNOT ring permutation (wire q <-> bit 3-q, MSB = wire 0).
  // Forward ring: CNOT(0,1), CNOT(1,2), CNOT(2,3), CNOT(3,0).
  int r = mr;
  r ^= (r & 1) << 3;          // undo CNOT(3,0)
  r ^= (r >> 1) & 1;          // undo CNOT(2,3)
  r ^= ((r >> 2) & 1) << 1;   // undo CNOT(1,2)
  r ^= ((r >> 3) & 1) << 2;   // undo CNOT(0,1)

  // U_RX[r][cr] = (prod_q cos|sin(w_q/2)) * (-i)^popcount(r^cr)
  float mag = 1.0f;
#pragma unroll
  for (int q = 0; q < 4; ++q) {
    float hw = 0.5f * w[l * 4 + q];
    int bit  = 3 - q;
    int diff = ((r >> bit) ^ (cr >> bit)) & 1;
    mag *= diff ? sinf(hw) : cosf(hw);
  }
  int d = __popc(r ^ cr) & 3;
  float re = (d == 0) ? mag : ((d == 2) ? -mag : 0.0f);
  float im = (d == 1) ? -mag : ((d == 3) ? mag : 0.0f);

  float val;
  if (mi == 0) val = (ci == 0) ? re : -im;        // [[Re, -Im],
  else         val = (ci == 0) ? im :  re;        //  [Im,  Re]]
  mats[(l * 32 + lane) * 32 + j] = val;
}

// ---------------------------------------------------------------------------
// Main: one wave simulates 16 samples. State held as 32x16 real matrix in
// WMMA B-operand layout (16 VGPRs): VGPR 2c+p holds rows {4c+p | 4c+p+2} in
// the two lane halves; rows 0-15 = Re(amp), 16-31 = Im(amp); N = lane & 15.
// ---------------------------------------------------------------------------
__global__ void __launch_bounds__(256) qdqn_main(
    const float* __restrict__ x, const float* __restrict__ mats,
    const float* __restrict__ Wm, const float* __restrict__ bvec,
    float* __restrict__ out, int B) {
  const int  lane = threadIdx.x & 31;
  const bool hi   = lane >= 16;
  const int  wave = blockIdx.x * (blockDim.x >> 5) + (threadIdx.x >> 5);
  const int  samp = wave * 16 + (lane & 15);
  const int  sidx = (samp < B) ? samp : (B - 1);

  // Per-sample RY half-angle cos/sin (data re-uploading uses same angles each layer).
  const float4 xv = reinterpret_cast<const float4*>(x)[sidx];
  float cc[4], ss[4];
  {
    const float invb[4] = {1.0f / 4.8f, 0.25f, 1.0f / 0.418f, 0.25f};
    const float xs[4]   = {xv.x, xv.y, xv.z, xv.w};
#pragma unroll
    for (int q = 0; q < 4; ++q) {
      float tq = xs[q] * invb[q];
      tq = fminf(fmaxf(tq, -1.0f), 1.0f);
      float h = tq * (0.5f * PI_F);               // (x_norm*pi)/2
      __sincosf(h, &ss[q], &cc[q]);
    }
  }
  const float s2sgn = hi ? ss[2] : -ss[2];        // qubit-2 cross-half butterfly sign

  float v[16];
#pragma unroll
  for (int i = 0; i < 16; ++i) v[i] = 0.0f;
  v[0] = hi ? 0.0f : 1.0f;                        // |0000>: Re(amp0)=1 (row 0, low half)

  for (int l = 0; l < 5; ++l) {
    // ---- per-sample RY embedding, butterflies in B-operand layout ----
#pragma unroll
    for (int g = 0; g < 16; g += 8)               // qubit 0 (MSB): VGPR pairs (i, i+4)
#pragma unroll
      for (int i = 0; i < 4; ++i) {
        float a = v[g + i], b = v[g + i + 4];
        v[g + i]     = cc[0] * a - ss[0] * b;
        v[g + i + 4] = ss[0] * a + cc[0] * b;
      }
    {
      const int q1i[4] = {0, 1, 4, 5};            // qubit 1: VGPR pairs (i, i+2)
#pragma unroll
      for (int g = 0; g < 16; g += 8)
#pragma unroll
        for (int t = 0; t < 4; ++t) {
          int i = q1i[t];
          float a = v[g + i], b = v[g + i + 2];
          v[g + i]     = cc[1] * a - ss[1] * b;
          v[g + i + 2] = ss[1] * a + cc[1] * b;
        }
    }
#pragma unroll
    for (int i = 0; i < 16; ++i) {                // qubit 2: lane-half pair via swizzle
      float sw = swapx16(v[i]);
      v[i] = cc[2] * v[i] + s2sgn * sw;
    }
#pragma unroll
    for (int g = 0; g < 16; g += 8)               // qubit 3 (LSB): VGPR pairs (i, i+1)
#pragma unroll
      for (int i = 0; i < 8; i += 2) {
        float a = v[g + i], b = v[g + i + 1];
        v[g + i]     = cc[3] * a - ss[3] * b;
        v[g + i + 1] = ss[3] * a + cc[3] * b;
      }

    // ---- fixed RX-ring + CNOT-ring unitary: 32x32 GEMM on the matrix core ----
    const float4* mp = reinterpret_cast<const float4*>(mats + ((l * 32 + lane) << 5));
    float a[32];
#pragma unroll
    for (int i = 0; i < 8; ++i) {
      float4 t = mp[i];
      a[4 * i + 0] = t.x; a[4 * i + 1] = t.y;
      a[4 * i + 2] = t.z; a[4 * i + 3] = t.w;
    }
    v8f dlo = {0.f, 0.f, 0.f, 0.f, 0.f, 0.f, 0.f, 0.f};
    v8f dhi = dlo;
#pragma unroll
    for (int k = 0; k < 8; ++k) {
      v2f bk;  bk.x  = v[2 * k];        bk.y  = v[2 * k + 1];
      v2f alo; alo.x = a[2 * k];        alo.y = a[2 * k + 1];
      v2f ahi; ahi.x = a[16 + 2 * k];   ahi.y = a[16 + 2 * k + 1];
      dlo = __builtin_amdgcn_wmma_f32_16x16x4_f32(false, alo, false, bk,
                                                  (short)0, dlo, false, false);
      dhi = __builtin_amdgcn_wmma_f32_16x16x4_f32(false, ahi, false, bk,
                                                  (short)0, dhi, false, false);
    }

    // ---- relayout C/D tile layout (VGPR r = rows {r | r+8}) back to B layout ----
    {
      float s0 = swapx16(dlo[0]), s1 = swapx16(dlo[1]), s2 = swapx16(dlo[2]), s3 = swapx16(dlo[3]);
      float s4 = swapx16(dlo[4]), s5 = swapx16(dlo[5]), s6 = swapx16(dlo[6]), s7 = swapx16(dlo[7]);
      v[0] = hi ? s2 : dlo[0];  v[1] = hi ? s3 : dlo[1];
      v[2] = hi ? s6 : dlo[4];  v[3] = hi ? s7 : dlo[5];
      v[4] = hi ? dlo[2] : s0;  v[5] = hi ? dlo[3] : s1;
      v[6] = hi ? dlo[6] : s4;  v[7] = hi ? dlo[7] : s5;
    }
    {
      float s0 = swapx16(dhi[0]), s1 = swapx16(dhi[1]), s2 = swapx16(dhi[2]), s3 = swapx16(dhi[3]);
      float s4 = swapx16(dhi[4]), s5 = swapx16(dhi[5]), s6 = swapx16(dhi[6]), s7 = swapx16(dhi[7]);
      v[8]  = hi ? s2 : dhi[0]; v[9]  = hi ? s3 : dhi[1];
      v[10] = hi ? s6 : dhi[4]; v[11] = hi ? s7 : dhi[5];
      v[12] = hi ? dhi[2] : s0; v[13] = hi ? dhi[3] : s1;
      v[14] = hi ? dhi[6] : s4; v[15] = hi ? dhi[7] : s5;
    }
  }

  // ---- probs -> <Z_q> -> linear head, folded: out_j = sum_i p_i * g_j(i) + b_j,
  //      g_j(i) = sum_q (+/-)W[j][q] by bit(3-q) of amp index i = 4c + p' + 2*half.
  float p[8];
#pragma unroll
  for (int i = 0; i < 8; ++i) p[i] = v[i] * v[i] + v[8 + i] * v[8 + i];

  float W0[4], W1[4];
#pragma unroll
  for (int q = 0; q < 4; ++q) { W0[q] = Wm[q]; W1[q] = Wm[4 + q]; }
  const float w02s = hi ? -W0[2] : W0[2];         // qubit-2 sign depends on lane half
  const float w12s = hi ? -W1[2] : W1[2];
  float acc0 = 0.0f, acc1 = 0.0f;
#pragma unroll
  for (int w = 0; w < 8; ++w) {
    const int   c   = w >> 1, pp = w & 1;
    const float sg0 = (c & 2) ? -1.0f : 1.0f;     // qubit 0 <- bit3 = c>>1
    const float sg1 = (c & 1) ? -1.0f : 1.0f;     // qubit 1 <- bit2 = c&1
    const float sg3 = pp ? -1.0f : 1.0f;          // qubit 3 <- bit0 = p'
    acc0 += p[w] * (sg0 * W0[0] + sg1 * W0[1] + sg3 * W0[3] + w02s);
    acc1 += p[w] * (sg0 * W1[0] + sg1 * W1[1] + sg3 * W1[3] + w12s);
  }
  acc0 += swapx16(acc0);                          // combine the two lane halves
  acc1 += swapx16(acc1);

  if (!hi && samp < B) {
    float2 o;
    o.x = acc0 + bvec[0];
    o.y = acc1 + bvec[1];
    reinterpret_cast<float2*>(out)[samp] = o;
  }
}

// ---------------------------------------------------------------------------
extern "C" void kernel_launch(void* const* d_in, const int* in_sizes, int n_in,
                              void* d_out, int out_size, void* d_ws, size_t ws_size,
                              hipStream_t stream) {
  const float* x       = (const float*)d_in[0];   // (B,4) f32
  const float* weights = (const float*)d_in[1];   // (5,4) f32
  const float* W       = (const float*)d_in[2];   // (2,4) f32
  const float* b       = (const float*)d_in[3];   // (2,)  f32
  float*       out     = (float*)d_out;           // (B,2) f32
  float*       mats    = (float*)d_ws;            // 5*32*32 f32 = 20 KB scratch

  const int B = in_sizes[0] / 4;

  qdqn_setup<<<20, 256, 0, stream>>>(weights, mats);

  const int waves  = (B + 15) / 16;               // 16 samples per wave
  const int blocks = (waves + 7) / 8;             // 8 waves (256 thr) per block
  qdqn_main<<<blocks, 256, 0, stream>>>(x, mats, W, b, out, B);
}